// diffeomorphic_transformation_37082747634495
// MI455X (gfx1250) — compile-verified
//
#include <hip/hip_runtime.h>

// ---------------------------------------------------------------------------
// Diffeomorphic (CPAB-style) image warp, MI455X / gfx1250.
//   N=32 batches, C=3, H=W=224, NX=NY=3 grid, 50 Euler steps.
// Stage 1: theta -> per-triangle affine maps A via V_WMMA_F32_16X16X4_F32
//          (full fp32 matrix op; M = 8 batches x 2 rows, K = 3 vertices,
//           N = 3 Pinv columns). Operands built branch-free in registers.
// Stage 2: per-pixel ODE integration (A table staged to LDS via the async
//          global->LDS path) + bilinear sampling.
// ---------------------------------------------------------------------------

#define NXC     3
#define NYC     3
#define NSTEPS  50
#define NBATCH  32
#define NCHAN   3
#define IMGH    224
#define IMGW    224
#define NPTS    (IMGH * IMGW)
#define DTHETA  26                       // 2*(4 interior) + 2*9 centers
#define AFLOATS (NYC * NXC * 4 * 2 * 3)  // 216 floats per batch

typedef float v2f __attribute__((ext_vector_type(2)));
typedef float v8f __attribute__((ext_vector_type(8)));

typedef __attribute__((address_space(1))) int as1_int;  // global
typedef __attribute__((address_space(3))) int as3_int;  // LDS

__device__ __forceinline__ float theta_val(const float* __restrict__ mean,
                                           const float* __restrict__ sig,
                                           const float* __restrict__ eps,
                                           int b, int i) {
    int idx = b * DTHETA + i;
    return eps[idx] * 0.1f * sig[idx] + mean[idx];
}

// ---------------------------------------------------------------------------
// Stage 1: one wave32 per (cell, triangle, batch-group of 8).
//   A[d][k] = sum_v U[v][d] * Pinv[v][k]
// WMMA 16x16x4 f32 mapping:
//   row M = batch_in_group*2 + d, K = vertex v (K=3 zero), col N = Pinv col k.
// Operand layouts per CDNA5 ISA 7.12.2:
//   A frag: lane l holds row M=l%16, VGPR0/1 = K { l<16 ? 0,1 : 2,3 }
//   B frag: VGPR0 lanes0-15=K0,lanes16-31=K2; VGPR1 lanes0-15=K1,lanes16-31=K3
//   D frag: VGPRj, lanes0-15 -> M=j, lanes16-31 -> M=j+8; N = lane%16
// ---------------------------------------------------------------------------
__global__ __launch_bounds__(32) void
compute_A_wmma(const float* __restrict__ mean, const float* __restrict__ sig,
               const float* __restrict__ eps, float* __restrict__ Aout) {
    const int ct   = blockIdx.x;   // cell*4 + tri, 0..35
    const int grp  = blockIdx.y;   // batch group of 8, 0..3
    const int cell = ct >> 2;
    const int tri  = ct & 3;
    const int cy = cell / NXC, cx = cell % NXC;

    const int lane = threadIdx.x;  // 0..31
    const int m    = lane & 15;    // row M
    const int half = lane >> 4;    // 0: K=0,1   1: K=2,3
    const int d    = m & 1;        // output row of the 2x3 affine
    const int b    = grp * 8 + (m >> 1);

    // triangle endpoint corner codes (0=p00,1=p10,2=p01,3=p11); 3rd vtx=center
    // t0:(0,1,c) t1:(1,3,c) t2:(0,2,c) t3:(2,3,c)   (block-uniform -> SALU)
    const int va = (tri == 1) ? 1 : ((tri == 3) ? 2 : 0);
    const int vb = (tri == 0) ? 1 : ((tri == 2) ? 2 : 3);

    // vertex positions for this cell
    const float xs0 = cx * (1.0f / 3.0f), xs1 = (cx + 1) * (1.0f / 3.0f);
    const float ys0 = cy * (1.0f / 3.0f), ys1 = (cy + 1) * (1.0f / 3.0f);
    const float ax  = (va & 1) ? xs1 : xs0, ay = (va >= 2) ? ys1 : ys0;
    const float bx  = (vb & 1) ? xs1 : xs0, by = (vb >= 2) ? ys1 : ys0;
    const float ccx = 0.5f * (xs0 + xs1),   ccy = 0.5f * (ys0 + ys1);

    // Pinv = inv([[ax,bx,ccx],[ay,by,ccy],[1,1,1]])  (block-uniform scalars)
    const float det = ax * (by - ccy) - bx * (ay - ccy) + ccx * (ay - by);
    const float id  = 1.0f / det;
    const float p00 = (by - ccy) * id, p01 = (ccx - bx) * id, p02 = (bx * ccy - ccx * by) * id;
    const float p10 = (ccy - ay) * id, p11 = (ax - ccx) * id, p12 = (ccx * ay - ax * ccy) * id;
    const float p20 = (ay - by) * id,  p21 = (bx - ax) * id,  p22 = (ax * by - bx * ay) * id;

    // vertex velocity for this lane's (batch b, row d)
    auto Ucorner = [&](int code) -> float {
        int ii = (code >= 2) ? cy + 1 : cy;   // corner grid row
        int jj = (code & 1) ? cx + 1 : cx;    // corner grid col
        float v = 0.0f;                       // boundary corners pinned to 0
        if (ii >= 1 && ii <= 2 && jj >= 1 && jj <= 2)
            v = theta_val(mean, sig, eps, b, (ii - 1) * 4 + (jj - 1) * 2 + d);
        return v;
    };
    const float Ucenter = theta_val(mean, sig, eps, b, 8 + (cy * 3 + cx) * 2 + d);

    // A operand: K = half*2 + vgpr; K==3 zero-padded
    v2f afrag;
    afrag.x = half ? Ucenter : Ucorner(va);
    afrag.y = half ? 0.0f    : Ucorner(vb);

    // B operand: column N = m; branchless Pinv[row][n] select
    const int n = m;
    auto rowsel = [&](float r0, float r1, float r2) -> float {
        float v = (n == 0) ? r0 : ((n == 1) ? r1 : r2);
        return (n < 3) ? v : 0.0f;
    };
    v2f bfrag;
    bfrag.x = half ? rowsel(p20, p21, p22) : rowsel(p00, p01, p02);
    bfrag.y = half ? 0.0f : rowsel(p10, p11, p12);

    v8f cz = { 0.0f, 0.0f, 0.0f, 0.0f, 0.0f, 0.0f, 0.0f, 0.0f };
    v8f dmat = __builtin_amdgcn_wmma_f32_16x16x4_f32(
        false, afrag, false, bfrag, (short)0, cz, false, false);

    // scatter D: lane's column n = Pinv column k; VGPR j + half -> row M
    if (n < 3) {
#pragma unroll
        for (int j = 0; j < 8; ++j) {
            int mr = half ? (8 + j) : j;
            int bo = grp * 8 + (mr >> 1);
            int dd = mr & 1;
            Aout[((bo * 9 + cell) * 4 + tri) * 6 + dd * 3 + n] = dmat[j];
        }
    }
}

// ---------------------------------------------------------------------------
// Stage 2: per-pixel 50-step Euler integration + bilinear sample.
// One block = 256 threads (8 wave32) on one batch-slice of pixels; the
// batch's 216-float A table is staged once into LDS (async global->LDS path)
// and reused 50x256 times.
// ---------------------------------------------------------------------------
#if __has_builtin(__builtin_amdgcn_global_load_async_to_lds_b32) && \
    __has_builtin(__builtin_amdgcn_s_wait_asynccnt)
#define USE_ASYNC_LDS 1
#else
#define USE_ASYNC_LDS 0
#endif

__global__ __launch_bounds__(256) void
warp_sample(const float* __restrict__ img, const float* __restrict__ A,
            float* __restrict__ out) {
    __shared__ float sA[AFLOATS];
    const int bnum = blockIdx.y;
    const int tid  = threadIdx.x;

#if USE_ASYNC_LDS
    if (tid < AFLOATS) {
        __builtin_amdgcn_global_load_async_to_lds_b32(
            (as1_int*)(A + bnum * AFLOATS + tid),
            (as3_int*)&sA[tid],
            /*offset=*/0, /*cpol=*/0);
    }
    __builtin_amdgcn_s_wait_asynccnt(0);
#else
    if (tid < AFLOATS) sA[tid] = A[bnum * AFLOATS + tid];
#endif
    __syncthreads();

    const int p = blockIdx.x * 256 + tid;
    if (p >= NPTS) return;
    const int iy = p / IMGW, jx = p % IMGW;

    float px = jx * (1.0f / (IMGW - 1));
    float py = iy * (1.0f / (IMGH - 1));
    const float dt = 1.0f / NSTEPS;

#pragma unroll 1
    for (int s = 0; s < NSTEPS; ++s) {
        int cx = (int)floorf(px * 3.0f); cx = cx < 0 ? 0 : (cx > 2 ? 2 : cx);
        int cy = (int)floorf(py * 3.0f); cy = cy < 0 ? 0 : (cy > 2 ? 2 : cy);
        float lx = px * 3.0f - (float)cx;
        float ly = py * 3.0f - (float)cy;
        int t = (ly <= lx) ? ((ly <= 1.0f - lx) ? 0 : 1)
                           : ((ly <= 1.0f - lx) ? 2 : 3);
        const float* At = &sA[((cy * 3 + cx) * 4 + t) * 6];
        float vx = At[0] * px + At[1] * py + At[2];
        float vy = At[3] * px + At[4] * py + At[5];
        px += dt * vx;
        py += dt * vy;
    }

    // bilinear sample with border clamp
    float gx = fminf(fmaxf(px, 0.0f), 1.0f) * (float)(IMGW - 1);
    float gy = fminf(fmaxf(py, 0.0f), 1.0f) * (float)(IMGH - 1);
    int x0 = (int)floorf(gx); x0 = x0 < 0 ? 0 : (x0 > IMGW - 2 ? IMGW - 2 : x0);
    int y0 = (int)floorf(gy); y0 = y0 < 0 ? 0 : (y0 > IMGH - 2 ? IMGH - 2 : y0);
    float wx = gx - (float)x0, wy = gy - (float)y0;
    float w00 = (1.0f - wx) * (1.0f - wy), w10 = wx * (1.0f - wy);
    float w01 = (1.0f - wx) * wy,          w11 = wx * wy;

#pragma unroll
    for (int c = 0; c < NCHAN; ++c) {
        const float* base =
            img + ((size_t)(bnum * NCHAN + c) * IMGH + y0) * IMGW + x0;
        float v = base[0] * w00 + base[1] * w10 +
                  base[IMGW] * w01 + base[IMGW + 1] * w11;
        out[((size_t)(bnum * NCHAN + c) * IMGH + iy) * IMGW + jx] = v;
    }
}

// ---------------------------------------------------------------------------
extern "C" void kernel_launch(void* const* d_in, const int* in_sizes, int n_in,
                              void* d_out, int out_size, void* d_ws,
                              size_t ws_size, hipStream_t stream) {
    const float* x    = (const float*)d_in[0];  // [32,3,224,224]
    const float* mean = (const float*)d_in[1];  // [32,26]
    const float* sig  = (const float*)d_in[2];  // [32,26]
    const float* eps  = (const float*)d_in[3];  // [32,26]
    float* A = (float*)d_ws;                    // [32,9,4,2,3] = 6912 floats

    compute_A_wmma<<<dim3(36, 4), 32, 0, stream>>>(mean, sig, eps, A);
    warp_sample<<<dim3((NPTS + 255) / 256, NBATCH), 256, 0, stream>>>(
        x, A, (float*)d_out);
}